// MIClassifier_24601572671793
// MI455X (gfx1250) — compile-verified
//
#include <hip/hip_runtime.h>
#include <hip/hip_bf16.h>
#include <math.h>

typedef float v2f __attribute__((ext_vector_type(2)));
typedef float v4f __attribute__((ext_vector_type(4)));
typedef float v8f __attribute__((ext_vector_type(8)));

#define N_INST 200000
#define DIM    768
#define TILES  (N_INST / 16)            // 12500 exactly
#define MAXBLK 512                      // blocks for blockmax
#define POOLROWS 1024                   // rows per pooling block
#define NPOOL ((N_INST + POOLROWS - 1) / POOLROWS)   // 196

// ---- workspace layout (bytes) ----
#define WS_BMAX    0u          // float[512]
#define WS_SCAL    2048u       // [0]=zmax, [1]=tau
#define WS_CAND    4096u       // float[200000]  (<=804096)
#define WS_SCORES  1048576u    // float[200000]  (<=1848576)
#define WS_PART    2097152u    // float[NPOOL*768] (<=2699264)

// ---------------------------------------------------------------------------
// Kernel 1: scores = x @ w_att + b_att using V_WMMA_F32_16X16X4_F32.
// One wave per 16-row tile. Per K-step of 4:
//   A (16x4 f32): lane l holds row M=l&15, K = 2*(l>>4)+{0,1}  -> float2 load
//   B (4x16 f32): lanes 0/16 hold w_att[k+2*(l>>4)+{0,1}], others 0
// Column N=0 of the 16x16 D accumulator = the 16 scores (lanes 0 and 16).
// ---------------------------------------------------------------------------
__global__ __launch_bounds__(256) void k_scores_wmma(
    const float* __restrict__ x, const float* __restrict__ w_att,
    const float* __restrict__ b_att, float* __restrict__ scores) {
  const int wave = threadIdx.x >> 5;
  const int lane = threadIdx.x & 31;
  const int tile = blockIdx.x * 8 + wave;
  if (tile >= TILES) return;                  // wave-uniform: EXEC stays full
  const int row0 = tile * 16;
  const int m    = lane & 15;
  const int sel  = lane >> 4;                 // 0 or 1
  const float* __restrict__ arow = x + (size_t)(row0 + m) * DIM + sel * 2;
  const float* __restrict__ wrow = w_att + sel * 2;
  const bool col0 = (m == 0);                 // lanes 0 and 16 carry B column 0

  v8f c = {0.f, 0.f, 0.f, 0.f, 0.f, 0.f, 0.f, 0.f};
#pragma unroll 4
  for (int k = 0; k < DIM; k += 4) {
    if ((k & 63) == 0) __builtin_prefetch(arow + k + 256, 0, 3);  // global_prefetch_b8
    v2f a  = *(const v2f*)(arow + k);         // 8B aligned
    v2f wv = *(const v2f*)(wrow + k);
    v2f b;
    b.x = col0 ? wv.x : 0.0f;
    b.y = col0 ? wv.y : 0.0f;
    c = __builtin_amdgcn_wmma_f32_16x16x4_f32(false, a, false, b,
                                              (short)0, c, false, false);
  }
  const float bias = b_att[0];
  if (lane == 0) {
#pragma unroll
    for (int v = 0; v < 8; ++v) scores[row0 + v] = c[v] + bias;      // M=0..7
  } else if (lane == 16) {
#pragma unroll
    for (int v = 0; v < 8; ++v) scores[row0 + 8 + v] = c[v] + bias;  // M=8..15
  }
}

// ---------------------------------------------------------------------------
// Kernel 2: per-block maxima of the score array.
// ---------------------------------------------------------------------------
__global__ __launch_bounds__(256) void k_blockmax(
    const float* __restrict__ scores, float* __restrict__ bmax) {
  __shared__ float red[256];
  float mv = -INFINITY;
  for (int i = blockIdx.x * 256 + threadIdx.x; i < N_INST; i += 256 * MAXBLK)
    mv = fmaxf(mv, scores[i]);
  red[threadIdx.x] = mv;
  __syncthreads();
  for (int s = 128; s > 0; s >>= 1) {
    if (threadIdx.x < s) red[threadIdx.x] = fmaxf(red[threadIdx.x], red[threadIdx.x + s]);
    __syncthreads();
  }
  if (threadIdx.x == 0) bmax[blockIdx.x] = red[0];
}

// ---------------------------------------------------------------------------
// Kernel 3 (single workgroup, 1024 threads):
//  (a) reduce block maxima -> zmax
//  (b) deterministic compaction of candidates { z > zmax-1 } via LDS scan
//  (c) bisection on g(tau)=sum relu(z-tau)=1 over candidates only
//  (d) exact tau = (S-1)/k over the identified support
// ---------------------------------------------------------------------------
__global__ __launch_bounds__(1024) void k_tau(
    const float* __restrict__ bmax, const float* __restrict__ scores,
    float* __restrict__ cand, float* __restrict__ scalars) {
  __shared__ float red[1024];
  __shared__ int   iscan[1024];
  __shared__ float sh_lo, sh_hi, sh_S;
  __shared__ int   sh_n;
  const int tid = threadIdx.x;

  // (a) zmax
  red[tid] = (tid < MAXBLK) ? bmax[tid] : -INFINITY;
  __syncthreads();
  for (int s = 512; s > 0; s >>= 1) {
    if (tid < s) red[tid] = fmaxf(red[tid], red[tid + s]);
    __syncthreads();
  }
  const float zmax = red[0];
  const float thr  = zmax - 1.0f;   // tau* in [thr, zmax]; z<=thr never in support
  __syncthreads();

  // (b) deterministic compaction (count -> exclusive scan -> write)
  int cnt = 0;
  for (int i = tid; i < N_INST; i += 1024) cnt += (scores[i] > thr) ? 1 : 0;
  iscan[tid] = cnt;
  __syncthreads();
  for (int s = 1; s < 1024; s <<= 1) {
    int v = (tid >= s) ? iscan[tid - s] : 0;
    __syncthreads();
    iscan[tid] += v;
    __syncthreads();
  }
  int off = iscan[tid] - cnt;       // exclusive prefix
  if (tid == 1023) sh_n = iscan[1023];
  if (tid == 0) { sh_lo = thr; sh_hi = zmax; }
  __syncthreads();
  for (int i = tid; i < N_INST; i += 1024) {
    float z = scores[i];
    if (z > thr) cand[off++] = z;
  }
  __syncthreads();                  // block-wide visibility of cand[]
  const int n = sh_n;

  // (c) bisection: invariant g(lo) >= 1 >= g(hi)
  for (int it = 0; it < 32; ++it) {
    const float mid = 0.5f * (sh_lo + sh_hi);
    float g = 0.f;
    for (int i = tid; i < n; i += 1024) g += fmaxf(cand[i] - mid, 0.f);
    red[tid] = g;
    __syncthreads();
    for (int s = 512; s > 0; s >>= 1) {
      if (tid < s) red[tid] += red[tid + s];
      __syncthreads();
    }
    if (tid == 0) { if (red[0] >= 1.0f) sh_lo = mid; else sh_hi = mid; }
    __syncthreads();
  }

  // (d) exact finish over support { z > lo }
  const float lo = sh_lo;
  float S = 0.f, K = 0.f;
  for (int i = tid; i < n; i += 1024) {
    float z = cand[i];
    if (z > lo) { S += z; K += 1.0f; }
  }
  red[tid] = S;
  __syncthreads();
  for (int s = 512; s > 0; s >>= 1) {
    if (tid < s) red[tid] += red[tid + s];
    __syncthreads();
  }
  if (tid == 0) sh_S = red[0];
  __syncthreads();
  red[tid] = K;
  __syncthreads();
  for (int s = 512; s > 0; s >>= 1) {
    if (tid < s) red[tid] += red[tid + s];
    __syncthreads();
  }
  if (tid == 0) {
    scalars[0] = zmax;
    scalars[1] = (sh_S - 1.0f) / red[0];   // tau (K >= 1 guaranteed)
  }
}

// ---------------------------------------------------------------------------
// Kernel 4: partial weighted pooling. 192 threads x 4 cols (float4) = 768.
// Weight w_i = relu(score_i - tau); rows with w==0 skip the 3KB x-row load
// entirely (support is tiny), so this pass costs ~scores-stream only.
// Per-block partials written unconditionally -> deterministic fixed-order
// reduction in k_final (no atomics).
// ---------------------------------------------------------------------------
__global__ __launch_bounds__(192) void k_pool(
    const float* __restrict__ x, const float* __restrict__ scores,
    const float* __restrict__ scalars, float* __restrict__ partial) {
  __shared__ float sw[192];
  const float tau = scalars[1];
  const int tid   = threadIdx.x;
  const int r0    = blockIdx.x * POOLROWS;
  const int nrows = min(POOLROWS, N_INST - r0);
  float acc0 = 0.f, acc1 = 0.f, acc2 = 0.f, acc3 = 0.f;

  for (int base = 0; base < nrows; base += 192) {
    const int cn = min(192, nrows - base);
    __syncthreads();
    float s = (tid < cn) ? scores[r0 + base + tid] : 0.f;
    sw[tid] = (tid < cn) ? fmaxf(s - tau, 0.f) : 0.f;
    __syncthreads();
    if ((base & 511) == 0) __builtin_prefetch(scores + r0 + base + 512, 0, 3);
    for (int r = 0; r < cn; ++r) {
      const float w = sw[r];
      if (w > 0.f) {                       // uniform across the block
        const float* row = x + (size_t)(r0 + base + r) * DIM + (tid << 2);
        v4f p = *(const v4f*)row;          // 16B aligned, fully coalesced
        acc0 += w * p.x; acc1 += w * p.y; acc2 += w * p.z; acc3 += w * p.w;
      }
    }
  }
  v4f o; o.x = acc0; o.y = acc1; o.z = acc2; o.w = acc3;
  *(v4f*)(partial + (size_t)blockIdx.x * DIM + (tid << 2)) = o;
}

// ---------------------------------------------------------------------------
// Kernel 5: reduce partials (fixed order), dot with w_cls, sigmoid.
// ---------------------------------------------------------------------------
__global__ __launch_bounds__(768) void k_final(
    const float* __restrict__ partial, const float* __restrict__ w_cls,
    const float* __restrict__ b_cls, float* __restrict__ out) {
  __shared__ float red[768];
  const int t = threadIdx.x;
  float a = 0.f;
  for (int b = 0; b < NPOOL; ++b) a += partial[(size_t)b * DIM + t];
  red[t] = a * w_cls[t];
  __syncthreads();
  if (t < 256) red[t] = red[t] + red[t + 256] + red[t + 512];
  __syncthreads();
  for (int s = 128; s > 0; s >>= 1) {
    if (t < s) red[t] += red[t + s];
    __syncthreads();
  }
  if (t == 0) out[0] = 1.0f / (1.0f + expf(-(red[0] + b_cls[0])));
}

// ---------------------------------------------------------------------------
extern "C" void kernel_launch(void* const* d_in, const int* in_sizes, int n_in,
                              void* d_out, int out_size, void* d_ws, size_t ws_size,
                              hipStream_t stream) {
  (void)in_sizes; (void)n_in; (void)out_size; (void)ws_size;
  const float* x     = (const float*)d_in[0];
  const float* w_att = (const float*)d_in[1];
  const float* b_att = (const float*)d_in[2];
  const float* w_cls = (const float*)d_in[3];
  const float* b_cls = (const float*)d_in[4];
  float* out = (float*)d_out;

  char* ws = (char*)d_ws;
  float* bmax    = (float*)(ws + WS_BMAX);
  float* scalars = (float*)(ws + WS_SCAL);
  float* cand    = (float*)(ws + WS_CAND);
  float* scores  = (float*)(ws + WS_SCORES);
  float* partial = (float*)(ws + WS_PART);

  k_scores_wmma<<<(TILES + 7) / 8, 256, 0, stream>>>(x, w_att, b_att, scores);
  k_blockmax<<<MAXBLK, 256, 0, stream>>>(scores, bmax);
  k_tau<<<1, 1024, 0, stream>>>(bmax, scores, cand, scalars);
  k_pool<<<NPOOL, 192, 0, stream>>>(x, scores, scalars, partial);
  k_final<<<1, 768, 0, stream>>>(partial, w_cls, b_cls, out);
}